// Nonlocal_diff_20255065768219
// MI455X (gfx1250) — compile-verified
//
#include <hip/hip_runtime.h>

typedef __attribute__((ext_vector_type(2))) float v2f;
typedef __attribute__((ext_vector_type(8))) float v8f;

#define WMMA_F32(acc, a, b)                                                    \
  acc = __builtin_amdgcn_wmma_f32_16x16x4_f32(false, a, false, b, (short)0,    \
                                              acc, false, false)

constexpr int   BB     = 8;
constexpr int   N      = 3136;      // 56*56
constexpr int   C      = 64;
constexpr int   Ci     = 32;
constexpr int   TPB    = N / 16;    // 196 row-tiles per batch
constexpr int   NTILES = BB * TPB;  // 1568
constexpr float LAM    = 0.1f;
constexpr float HSTEP  = 0.05f;
constexpr float BN_EPS = 1e-3f;
constexpr float INV_N  = 1.0f / (float)N;

// ---------------------------------------------------------------------------
// Kernel 1: theta = x@Wt+bt, phi = x@Wp+bp  (per-pixel 1x1 conv == GEMM),
// plus column sums of phi (per batch) for the f_sum term.
// One wave per 16-row tile. WMMA f32 16x16x4, K=64 -> 16 steps.
// ---------------------------------------------------------------------------
__global__ __launch_bounds__(32) void k_theta_phi(
    const float* __restrict__ x, const float* __restrict__ Wt,
    const float* __restrict__ bt, const float* __restrict__ Wp,
    const float* __restrict__ bp, float* __restrict__ theta,
    float* __restrict__ phi, float* __restrict__ sphi) {
  const int  tile    = blockIdx.x;
  const int  batch   = tile / TPB;
  const int  lane    = threadIdx.x;
  const int  col     = lane & 15;
  const int  koff    = (lane >> 4) * 2;   // K sub-offset for A/B fragments
  const int  rowsel  = (lane >> 4) * 8;   // C/D row group
  const long rowbase = (long)tile * 16;

  // A fragments: 16x64 x-tile, lane holds row (lane&15), K pair (koff,koff+1)
  v2f a[16];
#pragma unroll
  for (int kk = 0; kk < 16; ++kk) {
    const float* xp = x + (rowbase + col) * C + 4 * kk + koff;
    a[kk].x         = xp[0];
    a[kk].y         = xp[1];
  }

  v8f th0 = {}, th1 = {}, ph0 = {}, ph1 = {};
#pragma unroll
  for (int kk = 0; kk < 16; ++kk) {
    const int krow = 4 * kk + koff;
    v2f       w0, w1, p0, p1;
    w0.x = Wt[krow * Ci + col];
    w0.y = Wt[(krow + 1) * Ci + col];
    w1.x = Wt[krow * Ci + 16 + col];
    w1.y = Wt[(krow + 1) * Ci + 16 + col];
    p0.x = Wp[krow * Ci + col];
    p0.y = Wp[(krow + 1) * Ci + col];
    p1.x = Wp[krow * Ci + 16 + col];
    p1.y = Wp[(krow + 1) * Ci + 16 + col];
    WMMA_F32(th0, a[kk], w0);
    WMMA_F32(th1, a[kk], w1);
    WMMA_F32(ph0, a[kk], p0);
    WMMA_F32(ph1, a[kk], p1);
  }

  const float bt0 = bt[col], bt1 = bt[16 + col];
  const float bp0 = bp[col], bp1 = bp[16 + col];
  float       s0 = 0.f, s1 = 0.f;
#pragma unroll
  for (int v = 0; v < 8; ++v) {
    const int   row = v + rowsel;
    const float t0 = th0[v] + bt0, t1 = th1[v] + bt1;
    const float p0 = ph0[v] + bp0, p1 = ph1[v] + bp1;
    theta[(rowbase + row) * Ci + col]      = t0;
    theta[(rowbase + row) * Ci + 16 + col] = t1;
    phi[(rowbase + row) * Ci + col]        = p0;
    phi[(rowbase + row) * Ci + 16 + col]   = p1;
    s0 += p0;
    s1 += p1;
  }
  atomicAdd(&sphi[batch * Ci + col], s0);
  atomicAdd(&sphi[batch * Ci + 16 + col], s1);
}

// ---------------------------------------------------------------------------
// Kernel 2: M = phi^T @ g  per batch -> [32,64]; K = N = 3136 reduction.
// One block per batch; 8 waves = the 8 output 16x16 tiles.
// ---------------------------------------------------------------------------
__global__ __launch_bounds__(256) void k_reduce_M(
    const float* __restrict__ phi, const float* __restrict__ g,
    float* __restrict__ Mout) {
  const int batch  = blockIdx.x;
  const int wave   = threadIdx.x >> 5;
  const int lane   = threadIdx.x & 31;
  const int citile = wave >> 2;  // 0..1
  const int ctile  = wave & 3;   // 0..3
  const int m      = lane & 15;
  const int koff   = (lane >> 4) * 2;
  const int ci     = citile * 16 + m;  // A row (phi column index)
  const int cc     = ctile * 16 + m;   // B col (g channel)

  const float* pB = phi + (long)batch * N * Ci;
  const float* gB = g + (long)batch * N * C;

  v8f acc = {};
  for (int n0 = 0; n0 < N; n0 += 16) {
#pragma unroll
    for (int sub = 0; sub < 4; ++sub) {
      const int k0 = n0 + 4 * sub + koff;  // K index = pixel index
      v2f       a, b;
      a.x = pB[k0 * Ci + ci];        // A = phi^T: A[m][k] = phi[k][m]
      a.y = pB[(k0 + 1) * Ci + ci];
      b.x = gB[k0 * C + cc];         // B = g:    B[k][n] = g[k][n]
      b.y = gB[(k0 + 1) * C + cc];
      WMMA_F32(acc, a, b);
    }
  }
  const int rowsel = (lane >> 4) * 8;
#pragma unroll
  for (int v = 0; v < 8; ++v) {
    const int row = citile * 16 + v + rowsel;
    Mout[batch * (Ci * C) + row * C + ctile * 16 + (lane & 15)] = acc[v];
  }
}

// ---------------------------------------------------------------------------
// Kernel 3: per 16-row tile:
//   fg  = LAM * theta_tile(16x32) @ M(32x64)            (WMMA, K=32)
//   fgn = (fg - g_tile * f_sum) / N ; f_sum = LAM*theta.sphi
//   o   = fgn @ W_i + b_i ; BN ; ReLU                   (WMMA, K=64)
//   out = x + HSTEP * o
// 4 waves per block, wave j owns output column tile j (16 cols).
// ---------------------------------------------------------------------------
__global__ __launch_bounds__(128) void k_iter(
    const float* __restrict__ x, const float* __restrict__ theta,
    const float* __restrict__ sphi, const float* __restrict__ Mmat,
    const float* __restrict__ g, const float* __restrict__ Wi,
    const float* __restrict__ bi, const float* __restrict__ gam,
    const float* __restrict__ bet, const float* __restrict__ mmean,
    const float* __restrict__ mvar, float* __restrict__ out) {
  __shared__ float th_s[16 * Ci];
  __shared__ float g_s[16 * C];
  __shared__ float M_s[Ci * C];
  __shared__ float W_s[C * C];
  __shared__ float fgn_s[16 * C];
  __shared__ float fsum_s[16];
  __shared__ float sphi_s[Ci];

  const int  tile    = blockIdx.x;
  const int  batch   = tile / TPB;
  const int  tid     = threadIdx.x;
  const long rowbase = (long)tile * 16;

  for (int i = tid; i < 16 * Ci; i += 128) th_s[i] = theta[rowbase * Ci + i];
  for (int i = tid; i < 16 * C; i += 128) g_s[i] = g[rowbase * C + i];
  for (int i = tid; i < Ci * C; i += 128) M_s[i] = Mmat[batch * Ci * C + i];
  for (int i = tid; i < C * C; i += 128) W_s[i] = Wi[i];
  if (tid < Ci) sphi_s[tid] = sphi[batch * Ci + tid];
  __syncthreads();

  if (tid < 16) {
    float s = 0.f;
#pragma unroll
    for (int c = 0; c < Ci; ++c) s += th_s[tid * Ci + c] * sphi_s[c];
    fsum_s[tid] = LAM * s;  // f_sum for this row
  }
  __syncthreads();

  const int wave   = tid >> 5;  // output column tile 0..3
  const int lane   = tid & 31;
  const int r      = lane & 15;
  const int koff   = (lane >> 4) * 2;
  const int rowsel = (lane >> 4) * 8;
  const int colj   = wave * 16 + (lane & 15);

  // fg = theta @ M  (K = 32)
  v8f acc = {};
#pragma unroll
  for (int kk = 0; kk < 8; ++kk) {
    const int k = 4 * kk + koff;
    v2f       a, b;
    a.x = th_s[r * Ci + k];
    a.y = th_s[r * Ci + k + 1];
    b.x = M_s[k * C + colj];
    b.y = M_s[(k + 1) * C + colj];
    WMMA_F32(acc, a, b);
  }
#pragma unroll
  for (int v = 0; v < 8; ++v) {
    const int row = v + rowsel;
    const float fg = LAM * acc[v];
    fgn_s[row * C + colj] = (fg - g_s[row * C + colj] * fsum_s[row]) * INV_N;
  }
  __syncthreads();

  // o = fgn @ W_i  (K = 64)
  v8f acc2 = {};
#pragma unroll
  for (int kk = 0; kk < 16; ++kk) {
    const int k = 4 * kk + koff;
    v2f       a, b;
    a.x = fgn_s[r * C + k];
    a.y = fgn_s[r * C + k + 1];
    b.x = W_s[k * C + colj];
    b.y = W_s[(k + 1) * C + colj];
    WMMA_F32(acc2, a, b);
  }

  const float scale = gam[colj] * (1.0f / sqrtf(mvar[colj] + BN_EPS));
  const float shift = bet[colj] - mmean[colj] * scale;
  const float bias  = bi[colj];
#pragma unroll
  for (int v = 0; v < 8; ++v) {
    const int row = v + rowsel;
    float     o   = acc2[v] + bias;
    o             = o * scale + shift;
    o             = fmaxf(o, 0.f);
    out[(rowbase + row) * C + colj] =
        x[(rowbase + row) * C + colj] + HSTEP * o;
  }
}

// ---------------------------------------------------------------------------
extern "C" void kernel_launch(void* const* d_in, const int* in_sizes, int n_in,
                              void* d_out, int out_size, void* d_ws,
                              size_t ws_size, hipStream_t stream) {
  (void)in_sizes; (void)n_in; (void)out_size; (void)ws_size;
  const float* x      = (const float*)d_in[0];
  const float* Wt     = (const float*)d_in[1];
  const float* bt     = (const float*)d_in[2];
  const float* Wp     = (const float*)d_in[3];
  const float* bp     = (const float*)d_in[4];
  const float* Wstack = (const float*)d_in[5];
  const float* bstack = (const float*)d_in[6];
  const float* gamma  = (const float*)d_in[7];
  const float* beta   = (const float*)d_in[8];
  const float* mmean  = (const float*)d_in[9];
  const float* mvar   = (const float*)d_in[10];
  float*       out    = (float*)d_out;

  float* ws    = (float*)d_ws;
  float* theta = ws;                              // B*N*Ci
  float* phi   = theta + (size_t)BB * N * Ci;     // B*N*Ci
  float* sphi  = phi + (size_t)BB * N * Ci;       // B*Ci
  float* Mmat  = sphi + (size_t)BB * Ci;          // B*Ci*C
  // total ws: ~6.5 MB of fp32

  hipMemsetAsync(sphi, 0, (size_t)BB * Ci * sizeof(float), stream);
  k_theta_phi<<<NTILES, 32, 0, stream>>>(x, Wt, bt, Wp, bp, theta, phi, sphi);

  for (int i = 0; i < 2; ++i) {
    const float* gcur = (i == 0) ? x : (const float*)out;
    k_reduce_M<<<BB, 256, 0, stream>>>(phi, gcur, Mmat);
    k_iter<<<NTILES, 128, 0, stream>>>(
        x, theta, sphi, Mmat, gcur, Wstack + (size_t)i * C * C, bstack + i * C,
        gamma + i * C, beta + i * C, mmean + i * C, mvar + i * C, out);
  }
}